// IterativeVoxelModel_84507776516318
// MI455X (gfx1250) — compile-verified
//
#include <hip/hip_runtime.h>
#include <math.h>

// ---------------------------------------------------------------------------
// IterativeVoxelModel for MI455X (gfx1250, wave32, WMMA).
// All GEMMs use V_WMMA_F32_16X16X4_F32 (full fp32 precision, matches ref).
// Every row dimension is padded 125 -> 128 so all WMMA operand loads and
// stores are unconditional straight-line code (no exec-mask branches).
// Intermediates live in d_ws (~38MB) -> fully L2-resident (192MB L2).
// ---------------------------------------------------------------------------

#define NVOX   125
#define MP     128     // padded voxel rows
#define CH     256
#define NHD    8
#define HDIM   32
#define CTR    62      // center voxel index (2,2,2)
#define NBATCH 16

typedef float v2f __attribute__((ext_vector_type(2)));
typedef float v8f __attribute__((ext_vector_type(8)));

__device__ __forceinline__ v8f wmma4(v2f a, v2f b, v8f c) {
  // D = A(16x4,f32) x B(4x16,f32) + C(16x16,f32)
  return __builtin_amdgcn_wmma_f32_16x16x4_f32(false, a, false, b, (short)0, c,
                                               false, false);
}

__device__ __forceinline__ float gelu_exact(float x) {
  return 0.5f * x * (1.0f + erff(x * 0.70710678118654752f));
}

__device__ __forceinline__ float wave_sum_f(float v) {
  for (int o = 16; o > 0; o >>= 1) v += __shfl_xor(v, o, 32);
  return v;
}
__device__ __forceinline__ float wave_max_f(float v) {
  for (int o = 16; o > 0; o >>= 1) v = fmaxf(v, __shfl_xor(v, o, 32));
  return v;
}
__device__ __forceinline__ int wave_or_i(int v) {
  for (int o = 16; o > 0; o >>= 1) v |= __shfl_xor(v, o, 32);
  return v;
}

// ---------------------------------------------------------------------------
// x[b,v,c] = patch[b,v]*conv_w[c] + conv_b[c] + pos[v,c]; pad rows -> 0.
// grid: B*MP blocks, 256 threads
// ---------------------------------------------------------------------------
__global__ void embed_kernel(const float* __restrict__ patch,
                             const float* __restrict__ cw,
                             const float* __restrict__ cb,
                             const float* __restrict__ pos,
                             float* __restrict__ X) {
  const int row = blockIdx.x;          // b*MP + v
  const int b   = row >> 7;
  const int v   = row & (MP - 1);
  const int c   = threadIdx.x;
  float val = 0.0f;
  if (v < NVOX) {
    const float p = patch[b * NVOX + v];
    val = p * cw[c] + cb[c] + pos[(size_t)v * CH + c];
  }
  X[(size_t)row * CH + c] = val;
}

// ---------------------------------------------------------------------------
// LayerNorm over last dim (256). grid: B*MP blocks, 256 threads.
// (pad rows are normalized garbage -> finite; consumers discard them)
// ---------------------------------------------------------------------------
__global__ void ln_kernel(const float* __restrict__ X,
                          const float* __restrict__ g,
                          const float* __restrict__ bta,
                          float* __restrict__ Y) {
  const int row = blockIdx.x;
  const int c   = threadIdx.x;
  const float x = X[(size_t)row * CH + c];
  float s  = wave_sum_f(x);
  float s2 = wave_sum_f(x * x);
  __shared__ float sh[16];
  const int w = c >> 5, ln = c & 31;
  if (ln == 0) { sh[w] = s; sh[8 + w] = s2; }
  __syncthreads();
  float S = 0.f, S2 = 0.f;
  for (int i = 0; i < 8; ++i) { S += sh[i]; S2 += sh[8 + i]; }
  const float m    = S * (1.0f / 256.0f);
  const float var  = S2 * (1.0f / 256.0f) - m * m;
  const float inv  = rsqrtf(var + 1e-5f);
  Y[(size_t)row * CH + c] = (x - m) * inv * g[c] + bta[c];
}

// ---------------------------------------------------------------------------
// Generic per-batch GEMM: Y[b] = act(A[b](MPxK) @ W(KxN) + bias) [+ R[b]]
// One wave computes a 16 x (16*NT) tile; A fragment reused across NT WMMAs.
// grid: (B, N/(16*NT), MP/16), block: 32. No guards anywhere (padded M).
// ---------------------------------------------------------------------------
template <int ACT, int RES, int NT>
__global__ void gemm_kernel(const float* __restrict__ A,
                            const float* __restrict__ W,
                            const float* __restrict__ bias,
                            const float* R, float* Y,
                            int K, int N) {
  const int b    = blockIdx.x;
  const int ng   = blockIdx.y;
  const int mt   = blockIdx.z;
  const int lane = threadIdx.x;
  const int lo   = lane & 15;
  const int hi   = lane >> 4;
  const int row  = mt * 16 + lo;
  const int cb0  = ng * (16 * NT) + lo;
  const float* Ab = A + ((size_t)b * MP + row) * K;
  v8f acc[NT];
#pragma unroll
  for (int i = 0; i < NT; ++i) acc[i] = {};
#pragma unroll 4
  for (int k0 = 0; k0 < K; k0 += 4) {
    const int ka = k0 + 2 * hi;
    const v2f av = *reinterpret_cast<const v2f*>(Ab + ka);
    const float* Wk = W + (size_t)ka * N + cb0;
#pragma unroll
    for (int i = 0; i < NT; ++i) {
      v2f bv;
      bv[0] = Wk[16 * i];
      bv[1] = Wk[(size_t)N + 16 * i];
      acc[i] = wmma4(av, bv, acc[i]);
    }
  }
#pragma unroll
  for (int i = 0; i < NT; ++i) {
    const int col = cb0 + 16 * i;
    const float bcol = bias[col];
#pragma unroll
    for (int r = 0; r < 8; ++r) {
      const int m = mt * 16 + r + 8 * hi;
      float v = acc[i][r] + bcol;
      if (ACT == 1) v = gelu_exact(v);
      const size_t idx = ((size_t)b * MP + m) * N + col;
      if (RES) v += R[idx];
      Y[idx] = v;
    }
  }
}

// ---------------------------------------------------------------------------
// Attention scores over absolute grid positions:
//   S[b,v,h,n] = (1/sqrt(32)) * sum_d Q[b,v,h*32+d] * KG[b,n,h*32+d]
// S layout: (B, MP, NHD, MP). grid: (B, NHD, 16) [mt=z>>1, ng=z&1], block: 32
// ---------------------------------------------------------------------------
__global__ void attn_scores_kernel(const float* __restrict__ Q,
                                   const float* __restrict__ KG,
                                   float* __restrict__ S) {
  const int b    = blockIdx.x;
  const int h    = blockIdx.y;
  const int mt   = blockIdx.z >> 1;
  const int ng   = blockIdx.z & 1;
  const int lane = threadIdx.x;
  const int lo   = lane & 15;
  const int hi   = lane >> 4;
  const int row  = mt * 16 + lo;
  const float* Qb = Q  + ((size_t)b * MP + row) * CH + h * HDIM;
  const float* Kb = KG + (size_t)b * MP * CH + h * HDIM;
  v8f acc[4];
#pragma unroll
  for (int i = 0; i < 4; ++i) acc[i] = {};
#pragma unroll
  for (int k0 = 0; k0 < HDIM; k0 += 4) {
    const int ka = k0 + 2 * hi;
    const v2f av = *reinterpret_cast<const v2f*>(Qb + ka);
#pragma unroll
    for (int i = 0; i < 4; ++i) {
      const int col = ng * 64 + 16 * i + lo;
      v2f bv;
      bv[0] = Kb[(size_t)col * CH + ka];
      bv[1] = Kb[(size_t)col * CH + ka + 1];
      acc[i] = wmma4(av, bv, acc[i]);
    }
  }
  const float scale = 0.17677669529663687f;  // 32^-0.5
#pragma unroll
  for (int i = 0; i < 4; ++i) {
    const int col = ng * 64 + 16 * i + lo;
#pragma unroll
    for (int r = 0; r < 8; ++r) {
      const int m = mt * 16 + r + 8 * hi;
      S[(((size_t)b * MP + m) * NHD + h) * MP + col] = acc[i][r] * scale;
    }
  }
}

// ---------------------------------------------------------------------------
// Masked softmax in window space + scatter back to absolute positions.
// One block per (b, v); wave w == head h (wave32). Pad queries write zeros.
// Wf layout: (B, NHD, MP, MP); cW: (B, NHD, MP) padding-vb weight.
// ---------------------------------------------------------------------------
__global__ void attn_softmax_kernel(const float* __restrict__ S,
                                    const float* __restrict__ known,
                                    int layer,
                                    float* __restrict__ Wf,
                                    float* __restrict__ cW) {
  const int b    = blockIdx.x;
  const int v    = blockIdx.y;
  const int h    = threadIdx.x >> 5;
  const int lane = threadIdx.x & 31;
  float* Wrow = Wf + (((size_t)b * NHD + h) * MP + v) * MP;

  if (v >= NVOX) {  // padded query row: uniform per block, no divergence
#pragma unroll
    for (int t = 0; t < 4; ++t) Wrow[lane + 32 * t] = 0.f;
    if (lane == 0) cW[((size_t)b * NHD + h) * MP + v] = 0.f;
    return;
  }

  __shared__ float wsm[NHD][128];
  const int vx = v / 25, vy = (v / 5) % 5, vz = v % 5;
  const float* Srow = S + (((size_t)b * MP + v) * NHD + h) * MP;
  const bool post = (layer > 0) || (v >= CTR);

  float sv[4];
  int   mk[4];
  int   outc = 0;
#pragma unroll
  for (int t = 0; t < 4; ++t) {
    const int j = lane + 32 * t;      // window index
    sv[t] = 0.f; mk[t] = 0;
    if (j < NVOX) {
      const int ax = vx + j / 25 - 2;
      const int ay = vy + (j / 5) % 5 - 2;
      const int az = vz + j % 5 - 2;
      const bool inr = ((unsigned)ax < 5u) && ((unsigned)ay < 5u) && ((unsigned)az < 5u);
      if (!inr) outc++;
      const int alin = (ax * 5 + ay) * 5 + az;
      bool m = inr && (known[(size_t)b * NVOX + alin] > 0.5f) && (j != CTR) &&
               !(post && alin == CTR);
      mk[t] = m ? 1 : 0;
      if (m) sv[t] = Srow[alin];
    }
  }

  const int anym = wave_or_i(mk[0] | mk[1] | mk[2] | mk[3]);
  float wv[4];
  if (anym) {
    float mx = -3.0e38f;
#pragma unroll
    for (int t = 0; t < 4; ++t) if (mk[t]) mx = fmaxf(mx, sv[t]);
    mx = wave_max_f(mx);
    float s = 0.f;
#pragma unroll
    for (int t = 0; t < 4; ++t) {
      wv[t] = mk[t] ? __expf(sv[t] - mx) : 0.f;
      s += wv[t];
    }
    s = wave_sum_f(s);
    const float inv = 1.0f / s;
#pragma unroll
    for (int t = 0; t < 4; ++t) wv[t] *= inv;
  } else {
    // softmax of all-zero scores over every window slot: uniform 1/125.
#pragma unroll
    for (int t = 0; t < 4; ++t) wv[t] = (lane + 32 * t < NVOX) ? (1.0f / 125.0f) : 0.f;
  }
#pragma unroll
  for (int t = 0; t < 4; ++t) {
    const int j = lane + 32 * t;
    wsm[h][j] = (j < NVOX) ? wv[t] : 0.f;
  }
  // out-of-range window slots contribute vb (zero-padded grid) when !anym
  const float oc = wave_sum_f((float)outc);
  if (lane == 0)
    cW[((size_t)b * NHD + h) * MP + v] = anym ? 0.f : oc * (1.0f / 125.0f);
  __syncthreads();

  // Gather weights into absolute-position row (race-free: pure gather).
#pragma unroll
  for (int t = 0; t < 4; ++t) {
    const int n = lane + 32 * t;      // absolute key voxel (padded to 128)
    float val = 0.f;
    if (n < NVOX) {
      const int dx = n / 25 - vx, dy = (n / 5) % 5 - vy, dz = n % 5 - vz;
      if (dx >= -2 && dx <= 2 && dy >= -2 && dy <= 2 && dz >= -2 && dz <= 2) {
        const int j = ((dx + 2) * 5 + (dy + 2)) * 5 + (dz + 2);
        val = wsm[h][j];
      }
    }
    Wrow[n] = val;
  }
}

// ---------------------------------------------------------------------------
// O[b,v,h*32+d] = sum_n Wf[b,h,v,n] * VG[b,n,h*32+d] + cW[b,h,v]*vb[h*32+d]
// K=MP (Wf cols 125..127 are zero -> padded VG rows neutralized).
// grid: (B, NHD, MP/16), block: 32; NT=2 covers full 32-wide head.
// ---------------------------------------------------------------------------
__global__ void attn_out_kernel(const float* __restrict__ Wf,
                                const float* __restrict__ VG,
                                const float* __restrict__ cW,
                                const float* __restrict__ vbias,
                                float* __restrict__ O) {
  const int b    = blockIdx.x;
  const int h    = blockIdx.y;
  const int mt   = blockIdx.z;
  const int lane = threadIdx.x;
  const int lo   = lane & 15;
  const int hi   = lane >> 4;
  const int row  = mt * 16 + lo;
  const float* Ab = Wf + (((size_t)b * NHD + h) * MP + row) * MP;
  const float* Bb = VG + (size_t)b * MP * CH + h * HDIM + lo;
  v8f acc[2];
  acc[0] = {}; acc[1] = {};
#pragma unroll 4
  for (int k0 = 0; k0 < MP; k0 += 4) {
    const int ka = k0 + 2 * hi;
    const v2f av = *reinterpret_cast<const v2f*>(Ab + ka);
#pragma unroll
    for (int i = 0; i < 2; ++i) {
      v2f bv;
      bv[0] = Bb[(size_t)ka * CH + 16 * i];
      bv[1] = Bb[(size_t)(ka + 1) * CH + 16 * i];
      acc[i] = wmma4(av, bv, acc[i]);
    }
  }
#pragma unroll
  for (int i = 0; i < 2; ++i) {
    const int c = h * HDIM + 16 * i + lo;
    const float vb_c = vbias[c];
#pragma unroll
    for (int r = 0; r < 8; ++r) {
      const int m = mt * 16 + r + 8 * hi;
      const float cwv = cW[((size_t)b * NHD + h) * MP + m];
      O[((size_t)b * MP + m) * CH + c] = acc[i][r] + cwv * vb_c;
    }
  }
}

// ---------------------------------------------------------------------------
// out[b] = dot(X[b, CTR, :], head_w) + head_b
// ---------------------------------------------------------------------------
__global__ void head_kernel(const float* __restrict__ X,
                            const float* __restrict__ hw,
                            const float* __restrict__ hb,
                            float* __restrict__ out) {
  const int b = blockIdx.x;
  const int c = threadIdx.x;
  const float p = X[((size_t)b * MP + CTR) * CH + c] * hw[c];
  float s = wave_sum_f(p);
  __shared__ float sh[8];
  if ((c & 31) == 0) sh[c >> 5] = s;
  __syncthreads();
  if (c == 0) {
    float S = 0.f;
    for (int i = 0; i < 8; ++i) S += sh[i];
    out[b] = S + hb[0];
  }
}

// ---------------------------------------------------------------------------
extern "C" void kernel_launch(void* const* d_in, const int* in_sizes, int n_in,
                              void* d_out, int out_size, void* d_ws, size_t ws_size,
                              hipStream_t stream) {
  const float* patch  = (const float*)d_in[0];
  const float* known  = (const float*)d_in[1];
  const float* conv_w = (const float*)d_in[2];
  const float* conv_b = (const float*)d_in[3];
  const float* pos    = (const float*)d_in[4];
  const float* ln1_g  = (const float*)d_in[5];
  const float* ln1_b  = (const float*)d_in[6];
  const float* ln2_g  = (const float*)d_in[7];
  const float* ln2_b  = (const float*)d_in[8];
  const float* qW     = (const float*)d_in[9];
  const float* qb     = (const float*)d_in[10];
  const float* kW     = (const float*)d_in[11];
  const float* kb     = (const float*)d_in[12];
  const float* vW     = (const float*)d_in[13];
  const float* vb     = (const float*)d_in[14];
  const float* oW     = (const float*)d_in[15];
  const float* ob     = (const float*)d_in[16];
  const float* w1     = (const float*)d_in[17];
  const float* b1     = (const float*)d_in[18];
  const float* w2     = (const float*)d_in[19];
  const float* b2     = (const float*)d_in[20];
  const float* head_w = (const float*)d_in[21];
  const float* head_b = (const float*)d_in[22];
  float* out = (float*)d_out;

  // workspace partition (floats); all row dims padded to MP=128
  float* ws = (float*)d_ws;
  const size_t nBVC = (size_t)NBATCH * MP * CH;                 // 524288
  float* X   = ws;                 ws += nBVC;
  float* XN  = ws;                 ws += nBVC;
  float* Q   = ws;                 ws += nBVC;
  float* KG  = ws;                 ws += nBVC;
  float* VG  = ws;                 ws += nBVC;
  float* O   = ws;                 ws += nBVC;
  float* S   = ws;                 ws += (size_t)NBATCH * MP * NHD * MP;  // 2097152
  float* Wf  = ws;                 ws += (size_t)NBATCH * NHD * MP * MP;  // 2097152
  float* cWb = ws;                 ws += (size_t)NBATCH * NHD * MP;       // 16384
  float* H1  = ws;                 ws += (size_t)NBATCH * MP * 1024;      // 2097152
  (void)ws_size; (void)in_sizes; (void)n_in; (void)out_size;

  const dim3 rowGrid(NBATCH * MP);
  embed_kernel<<<rowGrid, CH, 0, stream>>>(patch, conv_w, conv_b, pos, X);

  for (int l = 0; l < 3; ++l) {
    const float* qW_l = qW + (size_t)l * CH * CH;
    const float* kW_l = kW + (size_t)l * CH * CH;
    const float* vW_l = vW + (size_t)l * CH * CH;
    const float* oW_l = oW + (size_t)l * CH * CH;
    const float* w1_l = w1 + (size_t)l * CH * 1024;
    const float* w2_l = w2 + (size_t)l * 1024 * CH;

    // q = LN1(x) @ qW + qb ; k/v projections use raw x (per reference)
    ln_kernel<<<rowGrid, CH, 0, stream>>>(X, ln1_g + l * CH, ln1_b + l * CH, XN);
    gemm_kernel<0, 0, 4><<<dim3(NBATCH, CH / 64, MP / 16), 32, 0, stream>>>(
        XN, qW_l, qb + l * CH, nullptr, Q, CH, CH);
    gemm_kernel<0, 0, 4><<<dim3(NBATCH, CH / 64, MP / 16), 32, 0, stream>>>(
        X, kW_l, kb + l * CH, nullptr, KG, CH, CH);
    gemm_kernel<0, 0, 4><<<dim3(NBATCH, CH / 64, MP / 16), 32, 0, stream>>>(
        X, vW_l, vb + l * CH, nullptr, VG, CH, CH);

    attn_scores_kernel<<<dim3(NBATCH, NHD, 16), 32, 0, stream>>>(Q, KG, S);
    attn_softmax_kernel<<<dim3(NBATCH, MP), 256, 0, stream>>>(S, known, l, Wf, cWb);
    attn_out_kernel<<<dim3(NBATCH, NHD, MP / 16), 32, 0, stream>>>(Wf, VG, cWb,
                                                                   vb + l * CH, O);

    // x = x + O @ oW + ob   (in-place residual)
    gemm_kernel<0, 1, 4><<<dim3(NBATCH, CH / 64, MP / 16), 32, 0, stream>>>(
        O, oW_l, ob + l * CH, X, X, CH, CH);

    // FFN: x = x + gelu(LN2(x) @ w1 + b1) @ w2 + b2
    ln_kernel<<<rowGrid, CH, 0, stream>>>(X, ln2_g + l * CH, ln2_b + l * CH, XN);
    gemm_kernel<1, 0, 4><<<dim3(NBATCH, 1024 / 64, MP / 16), 32, 0, stream>>>(
        XN, w1_l, b1 + l * 1024, nullptr, H1, CH, 1024);
    gemm_kernel<0, 1, 4><<<dim3(NBATCH, CH / 64, MP / 16), 32, 0, stream>>>(
        H1, w2_l, b2 + l * CH, X, X, 1024, CH);
  }

  head_kernel<<<NBATCH, CH, 0, stream>>>(X, head_w, head_b, out);
}